// QAGNNMultiHead_64596308131894
// MI455X (gfx1250) — compile-verified
//
#include <hip/hip_runtime.h>

#define N_NODES   50000
#define N_EDGES   1600000
#define HIDDEN    256
#define HALF_H    128
#define N_GRAPHS  1024
#define N_PROPS   8
#define IN_DIM    68

#define USE_ASYNC_STAGE 1

typedef float v2f __attribute__((ext_vector_type(2)));
typedef float v8f __attribute__((ext_vector_type(8)));

// ---------------------------------------------------------------------------
// Feature build: h0[n, 0:4]=float(x), [4:36]=emb24[x%24], [36:68]=emb72[x%72]
// ---------------------------------------------------------------------------
__global__ void build_features(const int* __restrict__ x,
                               const float* __restrict__ emb24,
                               const float* __restrict__ emb72,
                               float* __restrict__ h0) {
  int n = blockIdx.x * blockDim.x + threadIdx.x;
  if (n >= N_NODES) return;
  float* row = h0 + (long)n * IN_DIM;
  int xi[4];
#pragma unroll
  for (int j = 0; j < 4; ++j) { xi[j] = x[n * 4 + j]; row[j] = (float)xi[j]; }
#pragma unroll
  for (int j = 0; j < 4; ++j) {
    const float* e = emb24 + (xi[j] % 24) * 8;
#pragma unroll
    for (int k = 0; k < 8; ++k) row[4 + j * 8 + k] = e[k];
  }
#pragma unroll
  for (int j = 0; j < 4; ++j) {
    const float* e = emb72 + (xi[j] % 72) * 8;
#pragma unroll
    for (int k = 0; k < 8; ++k) row[36 + j * 8 + k] = e[k];
  }
}

// ---------------------------------------------------------------------------
// Fold h3=[b,e,b+e,b+2e] into qa weights: effw[h, 0:64]  = W0+W2+W3
//                                         effw[h,64:128] = W1+W2+2*W3
// ---------------------------------------------------------------------------
__global__ void build_effw(const float* __restrict__ qa_w, float* __restrict__ effw) {
  int t = blockIdx.x * blockDim.x + threadIdx.x;
  if (t >= HIDDEN * HALF_H) return;
  int h = t / HALF_H, k = t % HALF_H;
  const float* wr = qa_w + (long)h * HIDDEN;
  float v;
  if (k < 64) v = wr[k] + wr[128 + k] + wr[192 + k];
  else { int k2 = k - 64; v = wr[64 + k2] + wr[128 + k2] + 2.0f * wr[192 + k2]; }
  effw[t] = v;
}

// ---------------------------------------------------------------------------
// fp32 WMMA GEMM: C[M,N] = act(A[M,K] @ W[N,K]^T + bias), batched via blockIdx.z.
// Block = 8 waves; each wave owns a 32(M)x64(N) tile (2 A-frags x 4 B-frags,
// 8 v8f accumulators). The 64-row W strip (shared by all 8 waves) is staged
// per K-chunk into LDS via async global->LDS b128 (ASYNCcnt + barrier), with
// a 132-dword row stride (= 4 mod 64 -> the 4*l16 + 2*half b64 read pattern
// is bank-conflict-free).
// Tail handling: tile base is CLAMPED to the last valid 32-row pair (M is a
// multiple of 16, >= 32), so overshooting waves redundantly recompute and
// store identical values -> zero divergent guards, EXEC stays all-1s at
// every WMMA with purely straight-line inner-loop control flow.
// A fragment (16x4 f32): lane%16 = row; lanes0-15 k={kb,kb+1}, lanes16-31
// k={kb+2,kb+3}. B symmetric. C/D: VGPR r -> M=r+8*(lane/16), N=lane%16.
// ---------------------------------------------------------------------------
#define LSTR 132

template <int K, bool RELU, bool WRITE2X>
__global__ __launch_bounds__(256) void gemm_wmma_f32(
    const float* __restrict__ A, const float* __restrict__ W,
    const float* __restrict__ bias, float* __restrict__ C,
    float* __restrict__ C2x, int M, int N,
    long sA, long sW, long sB, long sC) {
  constexpr int CHUNK  = (K > 128) ? 128 : K;
  constexpr int NCHUNK = (K + CHUNK - 1) / CHUNK;
  __shared__ float bLds[64 * LSTR];

  const int tid  = threadIdx.x;
  const int lane = tid & 31;
  const int wid  = tid >> 5;
  const int z    = blockIdx.z;
  A += (long)z * sA; W += (long)z * sW; bias += (long)z * sB; C += (long)z * sC;

  const int nBase = blockIdx.y * 64;
  int mTile0 = (blockIdx.x * 8 + wid) * 2;
  const int mtMax = M / 16 - 2;      // last valid pair base; M % 16 == 0, M >= 32
  if (mTile0 > mtMax) mTile0 = mtMax;  // clamp: duplicate identical work/stores
  const int half = lane >> 4;
  const int l16  = lane & 15;

  v8f acc[2][4] = {};
  const float* aRow0 = A + (long)(mTile0 * 16 + l16) * K + 2 * half;
  const float* aRow1 = aRow0 + 16L * K;

#pragma unroll
  for (int ch = 0; ch < NCHUNK; ++ch) {
    const int k0 = ch * CHUNK;
    const int ck = (K - k0 < CHUNK) ? (K - k0) : CHUNK;
    // -- stage W[nBase .. nBase+63][k0 .. k0+ck) into LDS ------------------
#if USE_ASYNC_STAGE
    {
      const int nv = 64 * (ck / 4);              // float4 elements
      for (int idx = tid; idx < nv; idx += 256) {
        const int r = idx / (ck / 4);
        const int c = (idx - r * (ck / 4)) * 4;
        unsigned ldsOff = (unsigned)(unsigned long long)&bLds[r * LSTR + c];
        const float* gp = W + (long)(nBase + r) * K + k0 + c;
        asm volatile("global_load_async_to_lds_b128 %0, %1, off"
                     :: "v"(ldsOff), "v"(gp) : "memory");
      }
      asm volatile("s_wait_asynccnt 0x0" ::: "memory");
    }
#else
    for (int idx = tid; idx < 64 * ck; idx += 256) {
      int r = idx / ck, c = idx - r * ck;
      bLds[r * LSTR + c] = W[(long)(nBase + r) * K + k0 + c];
    }
#endif
    __syncthreads();
    // -- compute -----------------------------------------------------------
#pragma unroll
    for (int kb = 0; kb < ck; kb += 4) {
      v2f a0 = *(const v2f*)(aRow0 + k0 + kb);
      v2f a1 = *(const v2f*)(aRow1 + k0 + kb);
#pragma unroll
      for (int t = 0; t < 4; ++t) {
        v2f b = *(const v2f*)(&bLds[(t * 16 + l16) * LSTR + kb + 2 * half]);
        acc[0][t] = __builtin_amdgcn_wmma_f32_16x16x4_f32(
            false, a0, false, b, (short)0, acc[0][t], false, false);
        acc[1][t] = __builtin_amdgcn_wmma_f32_16x16x4_f32(
            false, a1, false, b, (short)0, acc[1][t], false, false);
      }
    }
    __syncthreads();
  }

  const float bv0 = bias[nBase + l16];
  const float bv1 = bias[nBase + 16 + l16];
  const float bv2 = bias[nBase + 32 + l16];
  const float bv3 = bias[nBase + 48 + l16];

#pragma unroll
  for (int mt = 0; mt < 2; ++mt) {
    const long mBase = (long)(mTile0 + mt) * 16;
#pragma unroll
    for (int r = 0; r < 8; ++r) {
      const long row  = mBase + r + half * 8;
      const long base = row * (long)N + nBase + l16;
      float v0 = acc[mt][0][r] + bv0;
      float v1 = acc[mt][1][r] + bv1;
      float v2 = acc[mt][2][r] + bv2;
      float v3 = acc[mt][3][r] + bv3;
      if (RELU) {
        v0 = fmaxf(v0, 0.f); v1 = fmaxf(v1, 0.f);
        v2 = fmaxf(v2, 0.f); v3 = fmaxf(v3, 0.f);
      }
      C[base] = v0; C[base + 16] = v1; C[base + 32] = v2; C[base + 48] = v3;
      if (WRITE2X) {
        C2x[base]      = 2.f * v0; C2x[base + 16] = 2.f * v1;
        C2x[base + 32] = 2.f * v2; C2x[base + 48] = 2.f * v3;
      }
    }
  }
}

// ---------------------------------------------------------------------------
// Edge scatter: aggr[dst] += L[src], 128 floats/edge, 1 wave per edge,
// float4 gather + 4 f32 atomics per lane. L and aggr (25.6 MB each) fit
// in the 192 MB L2, so the RMW stream stays on-chip.
// ---------------------------------------------------------------------------
__global__ void edge_scatter(const float* __restrict__ L,
                             const int* __restrict__ ei,
                             float* __restrict__ aggr) {
  long t    = (long)blockIdx.x * blockDim.x + threadIdx.x;
  int  e    = (int)(t >> 5);
  int  lane = (int)(t & 31);
  if (e >= N_EDGES) return;
  int src = ei[e];
  int dst = ei[N_EDGES + e];
  if (e + 64 < N_EDGES) {  // hint the gather stream (global_prefetch)
    int nsrc = ei[e + 64];
    __builtin_prefetch(L + (long)nsrc * HALF_H + lane * 4, 0, 0);
  }
  const float4 v = *(const float4*)(L + (long)src * HALF_H + lane * 4);
  float* d = aggr + (long)dst * HALF_H + lane * 4;
  atomicAdd(d + 0, v.x); atomicAdd(d + 1, v.y);
  atomicAdd(d + 2, v.z); atomicAdd(d + 3, v.w);
}

// ---------------------------------------------------------------------------
// Pooling
// ---------------------------------------------------------------------------
__global__ void zero_f32(float* __restrict__ p, int n) {
  int t = blockIdx.x * blockDim.x + threadIdx.x;
  if (t < n) p[t] = 0.f;
}

__global__ void pool_accum(const float* __restrict__ H,
                           const int* __restrict__ batch,
                           float* __restrict__ sums,
                           float* __restrict__ cnts) {
  int t = blockIdx.x * blockDim.x + threadIdx.x;
  if (t >= N_NODES * 64) return;
  int n  = t >> 6;
  int c4 = (t & 63) << 2;
  int b  = batch[n];
  const float4 v = *(const float4*)(H + (long)n * HIDDEN + c4);
  float* d = sums + (long)b * HIDDEN + c4;
  atomicAdd(d + 0, v.x); atomicAdd(d + 1, v.y);
  atomicAdd(d + 2, v.z); atomicAdd(d + 3, v.w);
  if ((t & 63) == 0) atomicAdd(cnts + b, 1.0f);
}

__global__ void pool_finalize(const float* __restrict__ sums,
                              const float* __restrict__ cnts,
                              float* __restrict__ g) {
  int t = blockIdx.x * blockDim.x + threadIdx.x;
  if (t >= N_GRAPHS * HIDDEN) return;
  g[t] = sums[t] / fmaxf(cnts[t / HIDDEN], 1.0f);
}

// ---------------------------------------------------------------------------
// Head layer 2: out[g,p] = h1[p,g,:] . w2[p,:] + b2[p]; one wave per (p,g)
// ---------------------------------------------------------------------------
__global__ void head2_kernel(const float* __restrict__ h1,
                             const float* __restrict__ w2,
                             const float* __restrict__ b2,
                             float* __restrict__ out) {
  int w    = (blockIdx.x * blockDim.x + threadIdx.x) >> 5;
  int lane = threadIdx.x & 31;
  if (w >= N_PROPS * N_GRAPHS) return;
  int p = w >> 10;            // / N_GRAPHS
  int g = w & (N_GRAPHS - 1);
  const float* hrow = h1 + ((long)p * N_GRAPHS + g) * HALF_H;
  const float* wrow = w2 + p * HALF_H;
  float s = 0.f;
#pragma unroll
  for (int i = 0; i < 4; ++i) s += hrow[lane + 32 * i] * wrow[lane + 32 * i];
  for (int off = 16; off; off >>= 1) s += __shfl_xor(s, off, 32);
  if (lane == 0) out[g * N_PROPS + p] = s + b2[p];
}

// ---------------------------------------------------------------------------
extern "C" void kernel_launch(void* const* d_in, const int* in_sizes, int n_in,
                              void* d_out, int out_size, void* d_ws, size_t ws_size,
                              hipStream_t stream) {
  const int*   x      = (const int*)d_in[0];
  const int*   ei     = (const int*)d_in[1];
  const int*   batch  = (const int*)d_in[2];
  const float* emb24  = (const float*)d_in[3];
  const float* emb72  = (const float*)d_in[4];
  const float* lin1_w = (const float*)d_in[5];
  const float* lin1_b = (const float*)d_in[6];
  const float* qa1_w  = (const float*)d_in[7];
  const float* qa1_b  = (const float*)d_in[8];
  const float* lin2_w = (const float*)d_in[9];
  const float* lin2_b = (const float*)d_in[10];
  const float* qa2_w  = (const float*)d_in[11];
  const float* qa2_b  = (const float*)d_in[12];
  const float* hw1    = (const float*)d_in[13];
  const float* hb1    = (const float*)d_in[14];
  const float* hw2    = (const float*)d_in[15];
  const float* hb2    = (const float*)d_in[16];
  float* out = (float*)d_out;
  (void)in_sizes; (void)n_in; (void)out_size; (void)ws_size;

  char* ws = (char*)d_ws;
  size_t off = 0;
  auto carve = [&](size_t bytes) -> float* {
    float* p = (float*)(ws + off);
    off = (off + bytes + 255) & ~(size_t)255;
    return p;
  };
  float* h0   = carve((size_t)N_NODES * IN_DIM * 4);   // 13.6 MB
  float* L    = carve((size_t)N_NODES * HALF_H * 4);   // 25.6 MB
  float* aggr = carve((size_t)N_NODES * HALF_H * 4);   // 25.6 MB
  float* H    = carve((size_t)N_NODES * HIDDEN * 4);   // 51.2 MB
  float* effw = carve((size_t)HIDDEN * HALF_H * 4);
  float* sums = carve((size_t)N_GRAPHS * HIDDEN * 4);
  float* cnts = carve((size_t)N_GRAPHS * 4);
  float* gp   = carve((size_t)N_GRAPHS * HIDDEN * 4);
  float* h1   = carve((size_t)N_PROPS * N_GRAPHS * HALF_H * 4); // 4 MB

  // 8 waves/block, 2 M-tiles (32 rows) per wave -> 256 rows per block-x step
  const int MTILES = N_NODES / 16;                       // 3125
  const int GX     = (MTILES + 15) / 16;                 // 196
  const int edgeBlocks = (int)(((long)N_EDGES * 32 + 255) / 256);

  build_features<<<(N_NODES + 255) / 256, 256, 0, stream>>>(x, emb24, emb72, h0);

  // ---- conv 1 ----
  build_effw<<<(HIDDEN * HALF_H + 255) / 256, 256, 0, stream>>>(qa1_w, effw);
  gemm_wmma_f32<IN_DIM, false, true><<<dim3(GX, HALF_H / 64, 1), 256, 0, stream>>>(
      h0, lin1_w, lin1_b, L, aggr, N_NODES, HALF_H, 0, 0, 0, 0);
  edge_scatter<<<edgeBlocks, 256, 0, stream>>>(L, ei, aggr);
  gemm_wmma_f32<HALF_H, true, false><<<dim3(GX, HIDDEN / 64, 1), 256, 0, stream>>>(
      aggr, effw, qa1_b, H, nullptr, N_NODES, HIDDEN, 0, 0, 0, 0);

  // ---- conv 2 ----
  build_effw<<<(HIDDEN * HALF_H + 255) / 256, 256, 0, stream>>>(qa2_w, effw);
  gemm_wmma_f32<HIDDEN, false, true><<<dim3(GX, HALF_H / 64, 1), 256, 0, stream>>>(
      H, lin2_w, lin2_b, L, aggr, N_NODES, HALF_H, 0, 0, 0, 0);
  edge_scatter<<<edgeBlocks, 256, 0, stream>>>(L, ei, aggr);
  gemm_wmma_f32<HALF_H, true, false><<<dim3(GX, HIDDEN / 64, 1), 256, 0, stream>>>(
      aggr, effw, qa2_b, H, nullptr, N_NODES, HIDDEN, 0, 0, 0, 0);

  // ---- global mean pool ----
  zero_f32<<<(N_GRAPHS * HIDDEN + 255) / 256, 256, 0, stream>>>(sums, N_GRAPHS * HIDDEN);
  zero_f32<<<(N_GRAPHS + 255) / 256, 256, 0, stream>>>(cnts, N_GRAPHS);
  pool_accum<<<(N_NODES * 64 + 255) / 256, 256, 0, stream>>>(H, batch, sums, cnts);
  pool_finalize<<<(N_GRAPHS * HIDDEN + 255) / 256, 256, 0, stream>>>(sums, cnts, gp);

  // ---- heads: batched WMMA GEMM (z = property) + wave-reduce layer 2 ----
  gemm_wmma_f32<HIDDEN, true, false>
      <<<dim3((N_GRAPHS / 16 + 15) / 16, HALF_H / 64, N_PROPS), 256, 0, stream>>>(
      gp, hw1, hb1, h1, nullptr, N_GRAPHS, HALF_H,
      0, (long)HALF_H * HIDDEN, (long)HALF_H, (long)N_GRAPHS * HALF_H);
  head2_kernel<<<(N_PROPS * N_GRAPHS * 32 + 255) / 256, 256, 0, stream>>>(h1, hw2, hb2, out);
}